// Attn_52974126629565
// MI455X (gfx1250) — compile-verified
//
#include <hip/hip_runtime.h>
#include <math.h>

#define B_SZ     32
#define S_SZ     1024
#define DEC_HID_ 1024
#define ENC2     1024      // 2*ENC_HID
#define ATTN_DIM_ 1024
#define M_TILE   64
#define LDS_STR  1028      // 1024 + 4-float pad -> conflict-free lane-strided reads

typedef __attribute__((ext_vector_type(2))) float v2f;
typedef __attribute__((ext_vector_type(8))) float v8f;
typedef __attribute__((ext_vector_type(4))) int   v4i;

#if defined(__gfx1250__) && __has_builtin(__builtin_amdgcn_global_load_async_to_lds_b128)
#define USE_ASYNC_LDS 1
typedef __attribute__((address_space(1))) v4i* gv4i_p;   // global int4*
typedef __attribute__((address_space(3))) v4i* lv4i_p;   // LDS int4*
#else
#define USE_ASYNC_LDS 0
#endif

__device__ __forceinline__ void wait_async_done() {
#if USE_ASYNC_LDS
#if __has_builtin(__builtin_amdgcn_s_wait_asynccnt)
    __builtin_amdgcn_s_wait_asynccnt(0);
#else
    asm volatile("s_wait_asynccnt 0x0" ::: "memory");
#endif
#endif
}

// ---------------------------------------------------------------------------
// Kernel 1: dec_proj[b][n] = decoder_hidden[b][:] @ W_dec[:, n] + bias[n]
// (0.1% of total FLOPs; W_dec rows read coalesced across n)
// ---------------------------------------------------------------------------
__global__ __launch_bounds__(256) void decproj_kernel(
    const float* __restrict__ dh, const float* __restrict__ W,
    const float* __restrict__ bias, float* __restrict__ decp)
{
    __shared__ float sdh[DEC_HID_];
    const int b = blockIdx.x;
    const int t = threadIdx.x;
    #pragma unroll
    for (int j = 0; j < 4; ++j) sdh[t + j * 256] = dh[b * DEC_HID_ + t + j * 256];
    __syncthreads();

    float a0 = 0.f, a1 = 0.f, a2 = 0.f, a3 = 0.f;
    for (int k = 0; k < DEC_HID_; ++k) {
        const float s = sdh[k];
        const float* wr = W + (size_t)k * ATTN_DIM_;
        a0 += s * wr[t];
        a1 += s * wr[t + 256];
        a2 += s * wr[t + 512];
        a3 += s * wr[t + 768];
    }
    float* o = decp + (size_t)b * ATTN_DIM_;
    o[t]       = a0 + bias[t];
    o[t + 256] = a1 + bias[t + 256];
    o[t + 512] = a2 + bias[t + 512];
    o[t + 768] = a3 + bias[t + 768];
}

// ---------------------------------------------------------------------------
// Kernel 2 (main): fused  scores[b][s] = v . tanh(enc[b,s,:] @ W_enc + dec_proj[b,:])
// One workgroup = (batch b, 64-row s-block). A-tile (64 x 1024 f32) staged in
// LDS via the CDNA5 async global->LDS path (ASYNCcnt) when available, full K.
// 8 waves each own 128 N columns; FP32 WMMA 16x16x4, one B fragment feeds 4
// M-subtiles. N is reduced in-register (tanh then dot with v) so enc_proj is
// never materialized in memory.
// ---------------------------------------------------------------------------
__global__ __launch_bounds__(256) void attn_scores_kernel(
    const float* __restrict__ enc,     // [B][S][1024]
    const float* __restrict__ W,       // [2048][1024]
    const float* __restrict__ decp,    // [B][1024]
    const float* __restrict__ v,       // [1024]
    float* __restrict__ scores)        // [B][S]
{
    extern __shared__ float sA[];                // [M_TILE][LDS_STR]
    __shared__ float sScores[M_TILE];

    const int b    = blockIdx.x;                 // 0..31
    const int s0   = blockIdx.y * M_TILE;        // 0..1023 step 64
    const int tid  = threadIdx.x;
    const int lane = tid & 31;
    const int wave = tid >> 5;                   // 0..7
    const int lanelo = lane & 15;
    const int koff   = (lane >> 4) << 1;         // 0 or 2 (A/B K sub-offset)

    // --- stage A-tile: enc[b][s0..s0+63][0..1023] into LDS -----------------
    const float4* src = (const float4*)(enc + ((size_t)b * S_SZ + s0) * ENC2);
#if USE_ASYNC_LDS
    // CDNA5 async copy: global -> LDS without touching VGPRs (ASYNCcnt path)
    #pragma unroll 4
    for (int j = 0; j < 64; ++j) {               // 16384 float4 / 256 threads
        const int idx = tid + j * 256;
        const int row = idx >> 8;                // /256 float4 per row
        const int c4  = idx & 255;
        const float* g = (const float*)(src + row * 256 + c4);
        float* l = sA + row * LDS_STR + (c4 << 2);
        __builtin_amdgcn_global_load_async_to_lds_b128((gv4i_p)g, (lv4i_p)l, 0, 0);
    }
    wait_async_done();                           // own wave's async writes done
#else
    #pragma unroll 4
    for (int j = 0; j < 64; ++j) {
        const int idx = tid + j * 256;
        const int row = idx >> 8;
        const int c4  = idx & 255;
        const float4 d = src[row * 256 + c4];
        float* dst = sA + row * LDS_STR + (c4 << 2);
        dst[0] = d.x; dst[1] = d.y; dst[2] = d.z; dst[3] = d.w;
    }
#endif
    if (tid < M_TILE) sScores[tid] = 0.0f;
    __syncthreads();

    const float* wEnc = W + (size_t)DEC_HID_ * ATTN_DIM_;   // W_enc block
    const float* dP   = decp + (size_t)b * ATTN_DIM_;

    // A-operand row pointers, one per 16-row M-subtile
    const float* aR0 = sA + ( 0 + lanelo) * LDS_STR + koff;
    const float* aR1 = sA + (16 + lanelo) * LDS_STR + koff;
    const float* aR2 = sA + (32 + lanelo) * LDS_STR + koff;
    const float* aR3 = sA + (48 + lanelo) * LDS_STR + koff;

    float part[4][8];
    #pragma unroll
    for (int m = 0; m < 4; ++m)
        #pragma unroll
        for (int i = 0; i < 8; ++i) part[m][i] = 0.0f;

    // wave owns N columns [wave*128, wave*128+128) -> 8 N-tiles of 16
    for (int t = 0; t < 8; ++t) {
        const int n0 = wave * 128 + t * 16;
        v8f C0 = {}, C1 = {}, C2 = {}, C3 = {};
        const float* wB = wEnc + (size_t)koff * ATTN_DIM_ + n0 + lanelo;

        for (int kk = 0; kk < 1024; kk += 4) {
            // B fragment: rows K=kk+koff, kk+koff+1 at column n0+lanelo
            const float* wp = wB + (size_t)kk * ATTN_DIM_;
            v2f bv; bv.x = wp[0]; bv.y = wp[ATTN_DIM_];
            // A fragments (LDS, conflict-free thanks to 1028-float stride)
            v2f a0, a1, a2, a3;
            a0.x = aR0[kk]; a0.y = aR0[kk + 1];
            a1.x = aR1[kk]; a1.y = aR1[kk + 1];
            a2.x = aR2[kk]; a2.y = aR2[kk + 1];
            a3.x = aR3[kk]; a3.y = aR3[kk + 1];
            C0 = __builtin_amdgcn_wmma_f32_16x16x4_f32(false, a0, false, bv, (short)0, C0, false, false);
            C1 = __builtin_amdgcn_wmma_f32_16x16x4_f32(false, a1, false, bv, (short)0, C1, false, false);
            C2 = __builtin_amdgcn_wmma_f32_16x16x4_f32(false, a2, false, bv, (short)0, C2, false, false);
            C3 = __builtin_amdgcn_wmma_f32_16x16x4_f32(false, a3, false, bv, (short)0, C3, false, false);
        }

        // epilogue for this N-tile: energy = tanh(C + dec_proj[n]); acc v[n]*energy
        const int n = n0 + lanelo;
        const float bias = dP[n];
        const float vn   = v[n];
        #pragma unroll
        for (int i = 0; i < 8; ++i) {
            part[0][i] += vn * tanhf(C0[i] + bias);
            part[1][i] += vn * tanhf(C1[i] + bias);
            part[2][i] += vn * tanhf(C2[i] + bias);
            part[3][i] += vn * tanhf(C3[i] + bias);
        }
    }

    // reduce over the 16 lanes of each half (sum over the wave's N columns)
    #pragma unroll
    for (int m = 0; m < 4; ++m)
        #pragma unroll
        for (int i = 0; i < 8; ++i) {
            float x = part[m][i];
            x += __shfl_xor(x, 1, 32);
            x += __shfl_xor(x, 2, 32);
            x += __shfl_xor(x, 4, 32);
            x += __shfl_xor(x, 8, 32);
            part[m][i] = x;
        }
    if (lanelo == 0) {
        const int half = lane >> 4;              // C VGPR i: rows i / i+8
        #pragma unroll
        for (int m = 0; m < 4; ++m)
            #pragma unroll
            for (int i = 0; i < 8; ++i)
                atomicAdd(&sScores[m * 16 + half * 8 + i], part[m][i]);
    }
    __syncthreads();
    if (tid < M_TILE) scores[(size_t)b * S_SZ + s0 + tid] = sScores[tid];
}

// ---------------------------------------------------------------------------
// Kernel 3: softmax over S=1024 per batch row
// ---------------------------------------------------------------------------
__global__ __launch_bounds__(256) void softmax_kernel(
    const float* __restrict__ scores, float* __restrict__ out)
{
    __shared__ float red[8];
    const int b = blockIdx.x;
    const int t = threadIdx.x;
    const int lane = t & 31, wave = t >> 5;
    const float* row = scores + (size_t)b * S_SZ;

    float x0 = row[t], x1 = row[t + 256], x2 = row[t + 512], x3 = row[t + 768];
    float m = fmaxf(fmaxf(x0, x1), fmaxf(x2, x3));
    for (int d = 16; d; d >>= 1) m = fmaxf(m, __shfl_xor(m, d, 32));
    if (lane == 0) red[wave] = m;
    __syncthreads();
    m = red[0];
    #pragma unroll
    for (int w = 1; w < 8; ++w) m = fmaxf(m, red[w]);
    __syncthreads();

    const float e0 = __expf(x0 - m), e1 = __expf(x1 - m);
    const float e2 = __expf(x2 - m), e3 = __expf(x3 - m);
    float s = e0 + e1 + e2 + e3;
    for (int d = 16; d; d >>= 1) s += __shfl_xor(s, d, 32);
    if (lane == 0) red[wave] = s;
    __syncthreads();
    s = 0.f;
    #pragma unroll
    for (int w = 0; w < 8; ++w) s += red[w];
    const float inv = 1.0f / s;

    float* o = out + (size_t)b * S_SZ;
    o[t]       = e0 * inv;
    o[t + 256] = e1 * inv;
    o[t + 512] = e2 * inv;
    o[t + 768] = e3 * inv;
}

// ---------------------------------------------------------------------------
extern "C" void kernel_launch(void* const* d_in, const int* in_sizes, int n_in,
                              void* d_out, int out_size, void* d_ws, size_t ws_size,
                              hipStream_t stream) {
    const float* dh   = (const float*)d_in[0];   // [32][1024]
    const float* enc  = (const float*)d_in[1];   // [32][1024][1024]
    const float* W    = (const float*)d_in[2];   // [2048][1024]
    const float* bias = (const float*)d_in[3];   // [1024]
    const float* v    = (const float*)d_in[4];   // [1024]
    float* out = (float*)d_out;                  // [32][1024]

    float* decp   = (float*)d_ws;                // 32*1024 f32
    float* scores = decp + B_SZ * ATTN_DIM_;     // 32*1024 f32

    decproj_kernel<<<B_SZ, 256, 0, stream>>>(dh, W, bias, decp);

    dim3 grid(B_SZ, S_SZ / M_TILE);              // 32 x 16 workgroups
    const size_t lds_bytes = (size_t)M_TILE * LDS_STR * sizeof(float); // ~257 KB (<320 KB WGP limit)
    attn_scores_kernel<<<grid, 256, lds_bytes, stream>>>(enc, W, decp, v, scores);

    softmax_kernel<<<B_SZ, 256, 0, stream>>>(scores, out);
}